// LIFNeuron_16286515987117
// MI455X (gfx1250) — compile-verified
//
#include <hip/hip_runtime.h>

// LIF neuron forward scan, T=8 time steps over dim 0.
//   mem   = (mem - spike)*DECAY + x[t]
//   spike = rint(clamp(mem, 0, 4))   (round-half-to-even, matching jnp.round)
//   out[t] = spike
//
// Memory-bound (206 MB streamed once, ~0.2 GFLOP): optimize purely for the
// gfx1250 memory path — b128 vector accesses, non-temporal hints (stream-once
// data must not pollute the 192 MB L2), all 8 independent loads issued before
// the register-resident scan.

#define LIF_T     8
#define LIF_DECAY 0.25f

typedef __attribute__((ext_vector_type(4))) float v4f;

__global__ __launch_bounds__(256) void lif_scan_v4(const float* __restrict__ x,
                                                   float* __restrict__ out,
                                                   int n4 /* = N/4 */) {
  const int i = blockIdx.x * blockDim.x + threadIdx.x;
  if (i >= n4) return;

  const v4f* __restrict__ xp = (const v4f*)x;
  v4f* __restrict__ op = (v4f*)out;

  // Issue all 8 time-step loads up-front (independent of the scan) so the
  // wave has 8 outstanding global_load_b128 NT requests before s_wait_loadcnt.
  v4f xv[LIF_T];
#pragma unroll
  for (int t = 0; t < LIF_T; ++t) {
    xv[t] = __builtin_nontemporal_load(&xp[t * n4 + i]);
  }

  v4f mem   = (v4f)0.0f;
  v4f spike = (v4f)0.0f;
#pragma unroll
  for (int t = 0; t < LIF_T; ++t) {
    mem = (mem - spike) * LIF_DECAY + xv[t];
    spike = __builtin_elementwise_rint(
        __builtin_elementwise_min(
            __builtin_elementwise_max(mem, (v4f)0.0f), (v4f)4.0f));
    __builtin_nontemporal_store(spike, &op[t * n4 + i]);
  }
}

// Scalar fallback (only used if the per-timestep element count isn't a
// multiple of 4 — not the case for the reference shape, but keeps the
// kernel shape-agnostic and deterministic).
__global__ __launch_bounds__(256) void lif_scan_scalar(const float* __restrict__ x,
                                                       float* __restrict__ out,
                                                       int n /* = N */) {
  const int i = blockIdx.x * blockDim.x + threadIdx.x;
  if (i >= n) return;

  float xv[LIF_T];
#pragma unroll
  for (int t = 0; t < LIF_T; ++t) {
    xv[t] = __builtin_nontemporal_load(&x[t * n + i]);
  }

  float mem = 0.0f, spike = 0.0f;
#pragma unroll
  for (int t = 0; t < LIF_T; ++t) {
    mem = (mem - spike) * LIF_DECAY + xv[t];
    spike = __builtin_rintf(__builtin_fminf(__builtin_fmaxf(mem, 0.0f), 4.0f));
    __builtin_nontemporal_store(spike, &out[t * n + i]);
  }
}

extern "C" void kernel_launch(void* const* d_in, const int* in_sizes, int n_in,
                              void* d_out, int out_size, void* d_ws, size_t ws_size,
                              hipStream_t stream) {
  const float* x = (const float*)d_in[0];
  float* out = (float*)d_out;

  const int ntotal = in_sizes[0] / LIF_T;  // spatial elements per time step

  if ((ntotal & 3) == 0) {
    const int n4 = ntotal >> 2;
    const int blocks = (n4 + 255) / 256;
    lif_scan_v4<<<blocks, 256, 0, stream>>>(x, out, n4);
  } else {
    const int blocks = (ntotal + 255) / 256;
    lif_scan_scalar<<<blocks, 256, 0, stream>>>(x, out, ntotal);
  }
}